// ModelNN2Layer_7834020348011
// MI455X (gfx1250) — compile-verified
//
#include <hip/hip_runtime.h>

typedef __attribute__((ext_vector_type(2))) float v2f;
typedef __attribute__((ext_vector_type(4))) float v4f;
typedef __attribute__((ext_vector_type(8))) float v8f;

#define NN 100000
#define NE 1600000
#define NG 512
#define F  128

#define AADD(p, v) __hip_atomic_fetch_add((p), (v), __ATOMIC_RELAXED, __HIP_MEMORY_SCOPE_AGENT)

// ---- degree / norm prep -------------------------------------------------
__global__ void init_ones(float* a, float* b, int n) {
    int i = blockIdx.x * blockDim.x + threadIdx.x;
    if (i < n) { a[i] = 1.0f; b[i] = 1.0f; }   // self-loop contributes 1 to both degrees
}

__global__ void deg_accum(const int* __restrict__ src, const int* __restrict__ dst,
                          float* degO, float* degI, int nE) {
    int e = blockIdx.x * blockDim.x + threadIdx.x;
    if (e < nE) {
        AADD(&degO[src[e]], 1.0f);
        AADD(&degI[dst[e]], 1.0f);
    }
}

__global__ void rsqrt_inplace(float* a, float* b, int n) {
    int i = blockIdx.x * blockDim.x + threadIdx.x;
    if (i < n) { a[i] = rsqrtf(a[i]); b[i] = rsqrtf(b[i]); }
}

__global__ void zero_f32(float* p, int n) {
    int i = blockIdx.x * blockDim.x + threadIdx.x;
    if (i < n) p[i] = 0.0f;
}

// ---- generic fp32 WMMA GEMM: D = post( A[MxK] @ W[KxN] ) ---------------
// One wave computes one 16x16 tile of D via V_WMMA_F32_16X16X4_F32.
// bias (len N) added if non-null; ReLU if doRelu; per-row scale if rowscale.
__global__ void gemm_wmma_f32(const float* __restrict__ A, const float* __restrict__ W,
                              const float* __restrict__ bias, const float* __restrict__ rowscale,
                              float* __restrict__ D, int M, int N, int K, int doRelu) {
    const int lane = threadIdx.x & 31;
    const int wave = threadIdx.x >> 5;
    const int tile = blockIdx.x * (blockDim.x >> 5) + wave;
    const int ntn  = N >> 4;
    const int row0 = (tile / ntn) << 4;
    const int col0 = (tile % ntn) << 4;
    if (row0 >= M) return;

    const int m    = lane & 15;      // A row / D col within tile
    const int hi   = lane >> 4;      // lane half selects K pair
    const int koff = hi << 1;

    const float* arow = A + (size_t)(row0 + m) * K + koff;
    const float* wcol = W + (size_t)koff * N + col0 + m;

    v8f acc = {};
    for (int kk = 0; kk < K; kk += 4) {
        v2f a = *(const v2f*)(arow + kk);            // A[m][kk+koff .. +1]
        v2f b;
        b.x = wcol[(size_t)kk * N];                  // W[kk+koff  ][col]
        b.y = wcol[(size_t)(kk + 1) * N];            // W[kk+koff+1][col]
        acc = __builtin_amdgcn_wmma_f32_16x16x4_f32(
            false, a, false, b, (short)0, acc, false, false);
    }

    const float bv = bias ? bias[col0 + m] : 0.0f;
#pragma unroll
    for (int r = 0; r < 8; ++r) {
        int row = row0 + r + (hi << 3);              // C: vgpr r -> row r + 8*hi
        float v = acc[r] + bv;
        if (doRelu)  v = fmaxf(v, 0.0f);
        if (rowscale) v *= rowscale[row];
        D[(size_t)row * N + col0 + m] = v;
    }
}

// ---- edge scatter: ACC[dst] += H[src]  (H already scaled by rsqrt(outdeg)) ----
// One wave per edge, each lane moves 4 floats.
__global__ void scatter_edges(const float* __restrict__ H, const int* __restrict__ src,
                              const int* __restrict__ dst, float* __restrict__ ACC, int nE) {
    unsigned t = blockIdx.x * blockDim.x + threadIdx.x;
    int e = (int)(t >> 5);
    if (e >= nE) return;
    int c = ((int)t & 31) << 2;
    int s = src[e], d = dst[e];
    v4f v = *(const v4f*)(H + (size_t)s * F + c);
    float* p = ACC + (size_t)d * F + c;
    AADD(p + 0, v.x);
    AADD(p + 1, v.y);
    AADD(p + 2, v.z);
    AADD(p + 3, v.w);
}

// ---- h = relu((acc + self) * rsqrt(indeg) + bias), in place over Hself ----
__global__ void finalize_layer(float* __restrict__ Hself, const float* __restrict__ ACCb,
                               const float* __restrict__ rsI, const float* __restrict__ bias,
                               int total) {
    int i = blockIdx.x * blockDim.x + threadIdx.x;
    if (i >= total) return;
    int n = i >> 7, k = i & (F - 1);
    Hself[i] = fmaxf((Hself[i] + ACCb[i]) * rsI[n] + bias[k], 0.0f);
}

// ---- per-graph sum readout with fused layer-2 activation ----------------
__global__ void readout(const float* __restrict__ ACCb, const float* __restrict__ Hself,
                        const float* __restrict__ rsI, const float* __restrict__ bias,
                        const int* __restrict__ gid, float* __restrict__ hg, int nNodes) {
    unsigned t = blockIdx.x * blockDim.x + threadIdx.x;
    int n = (int)(t >> 5);
    if (n >= nNodes) return;
    int c = ((int)t & 31) << 2;
    float rs = rsI[n];
    int g = gid[n];
    v4f a = *(const v4f*)(ACCb + (size_t)n * F + c);
    v4f h = *(const v4f*)(Hself + (size_t)n * F + c);
    float* p = hg + (size_t)g * F + c;
#pragma unroll
    for (int j = 0; j < 4; ++j) {
        float val = fmaxf((a[j] + h[j]) * rs + bias[c + j], 0.0f);
        AADD(p + j, val);
    }
}

// ---- final 128 -> 2 projection ------------------------------------------
__global__ void mlp_out(const float* __restrict__ a2, const float* __restrict__ fW3,
                        const float* __restrict__ fb3, float* __restrict__ out) {
    int t = blockIdx.x * blockDim.x + threadIdx.x;
    if (t >= NG * 2) return;
    int g = t >> 1, c = t & 1;
    float s = fb3[c];
    const float* row = a2 + (size_t)g * F;
    for (int k = 0; k < F; ++k) s += row[k] * fW3[k * 2 + c];
    out[t] = s;
}

extern "C" void kernel_launch(void* const* d_in, const int* in_sizes, int n_in,
                              void* d_out, int out_size, void* d_ws, size_t ws_size,
                              hipStream_t stream) {
    const float* x   = (const float*)d_in[0];
    const int*   src = (const int*)d_in[1];
    const int*   dst = (const int*)d_in[2];
    const int*   gid = (const int*)d_in[3];
    const float* W1  = (const float*)d_in[4];
    const float* b1  = (const float*)d_in[5];
    const float* W2  = (const float*)d_in[6];
    const float* b2  = (const float*)d_in[7];
    const float* fW1 = (const float*)d_in[8];
    const float* fb1 = (const float*)d_in[9];
    const float* fW2 = (const float*)d_in[10];
    const float* fb2 = (const float*)d_in[11];
    const float* fW3 = (const float*)d_in[12];
    const float* fb3 = (const float*)d_in[13];
    float* out = (float*)d_out;

    // workspace layout (~104 MB)
    float* BufA = (float*)d_ws;                         // 100000*128
    float* BufB = BufA + (size_t)NN * F;                // 100000*128
    float* rsO  = BufB + (size_t)NN * F;                // rsqrt(out_deg)
    float* rsI  = rsO + NN;                             // rsqrt(in_deg)
    float* hg   = rsI + NN;                             // 512*128
    float* a1   = hg + (size_t)NG * F;                  // 512*256
    float* a2m  = a1 + (size_t)NG * 256;                // 512*128

    const int B   = 256;
    const int NF  = NN * F;                             // 12,800,000

    // edge norms (self loops included via init=1)
    init_ones<<<(NN + B - 1) / B, B, 0, stream>>>(rsO, rsI, NN);
    deg_accum<<<(NE + B - 1) / B, B, 0, stream>>>(src, dst, rsO, rsI, NE);
    rsqrt_inplace<<<(NN + B - 1) / B, B, 0, stream>>>(rsO, rsI, NN);

    // ---- layer 1: H = (x @ W1) * rsO  -> BufA
    gemm_wmma_f32<<<(NN / 16) * (F / 16) / 8, B, 0, stream>>>(
        x, W1, nullptr, rsO, BufA, NN, F, F, 0);
    zero_f32<<<NF / B, B, 0, stream>>>(BufB, NF);
    scatter_edges<<<(NE * 32) / B, B, 0, stream>>>(BufA, src, dst, BufB, NE);
    // h1 = relu((acc + self)*rsI + b1), in place in BufA
    finalize_layer<<<NF / B, B, 0, stream>>>(BufA, BufB, rsI, b1, NF);

    // ---- layer 2: H = (h1 @ W2) * rsO -> BufB
    gemm_wmma_f32<<<(NN / 16) * (F / 16) / 8, B, 0, stream>>>(
        BufA, W2, nullptr, rsO, BufB, NN, F, F, 0);
    zero_f32<<<NF / B, B, 0, stream>>>(BufA, NF);
    scatter_edges<<<(NE * 32) / B, B, 0, stream>>>(BufB, src, dst, BufA, NE);

    // ---- readout: hg[g] += relu((acc + self)*rsI + b2)
    zero_f32<<<(NG * F) / B, B, 0, stream>>>(hg, NG * F);
    readout<<<(NN * 32) / B, B, 0, stream>>>(BufA, BufB, rsI, b2, gid, hg, NN);

    // ---- MLP head
    gemm_wmma_f32<<<(NG / 16) * (256 / 16) / 8, B, 0, stream>>>(
        hg, fW1, fb1, nullptr, a1, NG, 256, F, 1);
    gemm_wmma_f32<<<(NG / 16) * (F / 16) / 8, B, 0, stream>>>(
        a1, fW2, fb2, nullptr, a2m, NG, F, 256, 1);
    mlp_out<<<(NG * 2 + B - 1) / B, B, 0, stream>>>(a2m, fW3, fb3, out);
}